// NeRFRenderer_1657857376888
// MI455X (gfx1250) — compile-verified
//
#include <hip/hip_runtime.h>
#include <hip/hip_bf16.h>

typedef float v2f __attribute__((ext_vector_type(2)));
typedef float v8f __attribute__((ext_vector_type(8)));

#define WPB 4     // waves (rays) per block
#define NS 64     // coarse steps
#define NU 64     // upsample steps
#define NT 128    // merged samples per ray

struct RayScratch {
  float z_c[NS];
  float sig_c[NS];
  float zmid[NS];     // 63 used
  float cdf[NS];      // 63 used
  float z_f[NU];
  float sig_f[NU];
  float z_all[NT];
  float sig_all[NT];
  float w_all[NT];    // also used as temp for coarse weights
};

__device__ __forceinline__ float softplusf(float x) {
  return (x > 20.f) ? x : log1pf(expf(x));
}
__device__ __forceinline__ float sigmoidf(float x) {
  return 1.f / (1.f + expf(-x));
}

// Density MLP over a 16-sample tile of one ray.
// A (16x4, fp32): rows = samples; cols = [x, y, z, 1] (bias folded as homogeneous coord).
// B (4x16 per N-tile): rows = [W1_x, W1_y, W1_z, b1].
// Second layer 64->1 done as VALU dot + intra-half shfl reduction.
__device__ __forceinline__ void density_tile(
    const float* __restrict__ zarr, float* __restrict__ sig_out, int base,
    float ox, float oy, float oz, float dxr, float dyr, float dzr,
    const v2f bW[4], const float wsig[4], float bsig, int lane)
{
  const int m = lane & 15;
  const float zm = zarr[base + m];
  v2f a;
  if (lane < 16) { a.x = ox + dxr * zm; a.y = oy + dyr * zm; }  // K=0,1
  else           { a.x = oz + dzr * zm; a.y = 1.0f;          }  // K=2,3(bias)

  float p[8];
#pragma unroll
  for (int r = 0; r < 8; ++r) p[r] = 0.f;

#pragma unroll
  for (int t = 0; t < 4; ++t) {           // 4 N-tiles cover hidden=64
    v8f c = {};
    v8f h = __builtin_amdgcn_wmma_f32_16x16x4_f32(
        false, a, false, bW[t], (short)0, c, false, false);
#pragma unroll
    for (int r = 0; r < 8; ++r) {
      float hv = h[r];
      hv = hv > 0.f ? hv : 0.f;           // relu
      p[r] += hv * wsig[t];               // w_sigma[16t + (lane&15)]
    }
  }
  // D-layout: vgpr r holds row M=r (lanes 0-15) and M=r+8 (lanes 16-31).
#pragma unroll
  for (int r = 0; r < 8; ++r) {
    float v = p[r];
    v += __shfl_xor(v, 1);
    v += __shfl_xor(v, 2);
    v += __shfl_xor(v, 4);
    v += __shfl_xor(v, 8);                // reduce within each 16-lane half
    v = softplusf(v + bsig);
    if (lane == 0)  sig_out[base + r]     = v;
    if (lane == 16) sig_out[base + 8 + r] = v;
  }
}

// Color MLP over a 16-sample tile: input [x,y,z,dx,dy,dz,1,0] -> 2 chained K=4 WMMAs.
__device__ __forceinline__ void color_tile(
    const float* __restrict__ zarr, const float* __restrict__ warr, int base,
    float ox, float oy, float oz, float dxr, float dyr, float dzr,
    const v2f bC0[4], const v2f bC1[4],
    const float wcr[4], const float wcg[4], const float wcb[4],
    float b2r, float b2g, float b2b,
    float& accR, float& accG, float& accB, int lane)
{
  const int m = lane & 15;
  const float zm = zarr[base + m];
  v2f a0, a1;
  if (lane < 16) { a0.x = ox + dxr * zm; a0.y = oy + dyr * zm;   // K=0,1: x,y
                   a1.x = dyr;           a1.y = dzr; }           // K=4,5: dy,dz
  else           { a0.x = oz + dzr * zm; a0.y = dxr;             // K=2,3: z,dx
                   a1.x = 1.f;           a1.y = 0.f; }           // K=6(bias),7(pad)

  float pr[8], pg[8], pb[8];
#pragma unroll
  for (int r = 0; r < 8; ++r) { pr[r] = 0.f; pg[r] = 0.f; pb[r] = 0.f; }

#pragma unroll
  for (int t = 0; t < 4; ++t) {
    v8f c = {};
    c = __builtin_amdgcn_wmma_f32_16x16x4_f32(
        false, a0, false, bC0[t], (short)0, c, false, false);
    c = __builtin_amdgcn_wmma_f32_16x16x4_f32(          // chained K accumulation
        false, a1, false, bC1[t], (short)0, c, false, false);
#pragma unroll
    for (int r = 0; r < 8; ++r) {
      float hv = c[r];
      hv = hv > 0.f ? hv : 0.f;
      pr[r] += hv * wcr[t];
      pg[r] += hv * wcg[t];
      pb[r] += hv * wcb[t];
    }
  }
#pragma unroll
  for (int r = 0; r < 8; ++r) {
    float vr = pr[r], vg = pg[r], vb = pb[r];
    vr += __shfl_xor(vr, 1); vr += __shfl_xor(vr, 2); vr += __shfl_xor(vr, 4); vr += __shfl_xor(vr, 8);
    vg += __shfl_xor(vg, 1); vg += __shfl_xor(vg, 2); vg += __shfl_xor(vg, 4); vg += __shfl_xor(vg, 8);
    vb += __shfl_xor(vb, 1); vb += __shfl_xor(vb, 2); vb += __shfl_xor(vb, 4); vb += __shfl_xor(vb, 8);
    if (lane == 0 || lane == 16) {
      int mm = base + ((lane >> 4) << 3) + r;
      float w = warr[mm];
      accR += w * sigmoidf(vr + b2r);
      accG += w * sigmoidf(vg + b2g);
      accB += w * sigmoidf(vb + b2b);
    }
  }
}

__global__ __launch_bounds__(WPB * 32) void nerf_fused_kernel(
    const float* __restrict__ rays_o, const float* __restrict__ rays_d,
    const float* __restrict__ W1, const float* __restrict__ b1,
    const float* __restrict__ w_sigma, const float* __restrict__ b_sigma,
    const float* __restrict__ Wc1, const float* __restrict__ bc1,
    const float* __restrict__ Wc2, const float* __restrict__ bc2,
    float* __restrict__ out, int R)
{
  __shared__ float sW1[3 * 64];
  __shared__ float sB1[64];
  __shared__ float sWs[64];
  __shared__ float sWc1[6 * 64];
  __shared__ float sBc1[64];
  __shared__ float sWc2[64 * 3];
  __shared__ float sBc2[3];
  __shared__ RayScratch rs[WPB];

  const int tid = threadIdx.x;
  for (int i = tid; i < 3 * 64; i += blockDim.x) sW1[i] = W1[i];
  for (int i = tid; i < 64;     i += blockDim.x) { sB1[i] = b1[i]; sWs[i] = w_sigma[i]; sBc1[i] = bc1[i]; }
  for (int i = tid; i < 6 * 64; i += blockDim.x) sWc1[i] = Wc1[i];
  for (int i = tid; i < 64 * 3; i += blockDim.x) sWc2[i] = Wc2[i];
  if (tid < 3) sBc2[tid] = bc2[tid];
  __syncthreads();

  const int lane = tid & 31;
  const int wv = tid >> 5;
  RayScratch& S = rs[wv];

  const int rid = blockIdx.x * WPB + wv;
  const bool valid = rid < R;
  const int ray = valid ? rid : (R - 1);

  const float bsig = b_sigma[0];
  const float b2r = sBc2[0], b2g = sBc2[1], b2b = sBc2[2];

  // ---- Build per-lane WMMA B fragments (4x16 tiles; bias folded as last K-row) ----
  const int half = lane >> 4;
  const int c16 = lane & 15;
  v2f bW[4], bC0[4], bC1[4];
  float wsig[4], wcr[4], wcg[4], wcb[4];
#pragma unroll
  for (int t = 0; t < 4; ++t) {
    const int col = t * 16 + c16;
    if (half == 0) {
      bW[t].x  = sW1[0 * 64 + col]; bW[t].y  = sW1[1 * 64 + col];
      bC0[t].x = sWc1[0 * 64 + col]; bC0[t].y = sWc1[1 * 64 + col];
      bC1[t].x = sWc1[4 * 64 + col]; bC1[t].y = sWc1[5 * 64 + col];
    } else {
      bW[t].x  = sW1[2 * 64 + col]; bW[t].y  = sB1[col];
      bC0[t].x = sWc1[2 * 64 + col]; bC0[t].y = sWc1[3 * 64 + col];
      bC1[t].x = sBc1[col];          bC1[t].y = 0.f;
    }
    wsig[t] = sWs[col];
    wcr[t] = sWc2[col * 3 + 0];
    wcg[t] = sWc2[col * 3 + 1];
    wcb[t] = sWc2[col * 3 + 2];
  }

  // ---- Per-ray data (uniform across wave) ----
  const float ox = rays_o[ray * 3 + 0], oy = rays_o[ray * 3 + 1], oz = rays_o[ray * 3 + 2];
  const float dxr = rays_d[ray * 3 + 0], dyr = rays_d[ray * 3 + 1], dzr = rays_d[ray * 3 + 2];

  // ---- near/far against [-1,1]^3 ----
  float tnear, tfar;
  {
    float t0x = (-1.f - ox) / (dxr + 1e-15f), t1x = (1.f - ox) / (dxr + 1e-15f);
    float t0y = (-1.f - oy) / (dyr + 1e-15f), t1y = (1.f - oy) / (dyr + 1e-15f);
    float t0z = (-1.f - oz) / (dzr + 1e-15f), t1z = (1.f - oz) / (dzr + 1e-15f);
    float nx = fminf(t0x, t1x), fx = fmaxf(t0x, t1x);
    float ny = fminf(t0y, t1y), fy = fmaxf(t0y, t1y);
    float nz = fminf(t0z, t1z), fz = fmaxf(t0z, t1z);
    tnear = fmaxf(nx, fmaxf(ny, nz));
    tfar  = fminf(fx, fminf(fy, fz));
    if (tfar < tnear) { tnear = 1e9f; tfar = 1e9f; }
    tnear = fmaxf(tnear, 0.5f);
  }
  const float sample_dist = (tfar - tnear) / (float)NS;

  // ---- coarse z (linspace over [near,far], 64 pts) ----
  {
    const int i0 = lane * 2, i1 = i0 + 1;
    S.z_c[i0] = tnear + (tfar - tnear) * ((float)i0 * (1.f / 63.f));
    S.z_c[i1] = tnear + (tfar - tnear) * ((float)i1 * (1.f / 63.f));
  }
  __syncthreads();

  // ---- coarse density (4 WMMA tiles x 4 N-tiles) ----
  for (int b = 0; b < NS; b += 16)
    density_tile(S.z_c, S.sig_c, b, ox, oy, oz, dxr, dyr, dzr, bW, wsig, bsig, lane);
  __syncthreads();

  // ---- coarse compositing weights via wave prefix-product; z_mid ----
  {
    const int i0 = lane * 2, i1 = i0 + 1;
    const float z0 = S.z_c[i0], z1 = S.z_c[i1];
    const float zn = (i1 < 63) ? S.z_c[i1 + 1] : z1;
    const float d0 = z1 - z0;
    const float d1 = (i1 == 63) ? sample_dist : (zn - z1);
    const float a0 = 1.f - expf(-d0 * S.sig_c[i0]);
    const float a1 = 1.f - expf(-d1 * S.sig_c[i1]);
    const float q0 = 1.f - a0 + 1e-15f;
    const float q1 = 1.f - a1 + 1e-15f;
    float inc = q0 * q1;
#pragma unroll
    for (int dlt = 1; dlt < 32; dlt <<= 1) {
      float up = __shfl_up(inc, dlt);
      if (lane >= dlt) inc *= up;
    }
    float exc = __shfl_up(inc, 1);
    if (lane == 0) exc = 1.f;
    S.w_all[i0] = a0 * exc;            // temp: coarse weights
    S.w_all[i1] = a1 * exc * q0;
    S.zmid[i0] = z0 + 0.5f * d0;
    if (i1 < 63) S.zmid[i1] = z1 + 0.5f * d1;
  }
  __syncthreads();

  // ---- CDF over w_coarse[1:-1] (62 vals) via wave prefix-sum ----
  {
    const int k0 = lane * 2, k1 = k0 + 1;
    float wa = (k0 <= 61) ? (S.w_all[k0 + 1] + 1e-5f) : 0.f;
    float wb = (k1 <= 61) ? (S.w_all[k1 + 1] + 1e-5f) : 0.f;
    float inc = wa + wb;
#pragma unroll
    for (int dlt = 1; dlt < 32; dlt <<= 1) {
      float up = __shfl_up(inc, dlt);
      if (lane >= dlt) inc += up;
    }
    const float total = __shfl(inc, 31);
    const float inv = 1.f / total;
    float exc = __shfl_up(inc, 1);
    if (lane == 0) { exc = 0.f; S.cdf[0] = 0.f; }
    if (k0 <= 61) S.cdf[k0 + 1] = (exc + wa) * inv;
    if (k1 <= 61) S.cdf[k1 + 1] = (exc + wa + wb) * inv;
  }
  __syncthreads();

  // ---- deterministic inverse-CDF sampling (searchsorted right, 2 u's/lane) ----
#pragma unroll
  for (int kk = 0; kk < 2; ++kk) {
    const int k = lane + kk * 32;
    const float u = ((float)k + 0.5f) * (1.f / 64.f);
    int lo = 0, hi = 63;
    while (lo < hi) { int mid = (lo + hi) >> 1; if (S.cdf[mid] <= u) lo = mid + 1; else hi = mid; }
    int below = lo - 1; if (below < 0) below = 0;
    int above = lo;     if (above > 62) above = 62;
    const float c0 = S.cdf[below], c1 = S.cdf[above];
    const float g0 = S.zmid[below], g1 = S.zmid[above];
    float den = c1 - c0; if (den < 1e-5f) den = 1.f;
    S.z_f[k] = g0 + ((u - c0) / den) * (g1 - g0);
  }
  __syncthreads();

  // ---- fine density ----
  for (int b = 0; b < NU; b += 16)
    density_tile(S.z_f, S.sig_f, b, ox, oy, oz, dxr, dyr, dzr, bW, wsig, bsig, lane);
  __syncthreads();

  // ---- stable merge of two sorted 64-lists (rank via binary search) ----
#pragma unroll
  for (int kk = 0; kk < 2; ++kk) {
    const int i = lane + kk * 32;
    {
      const float v = S.z_c[i];
      int lo = 0, hi = 64;                       // count of z_f strictly < v
      while (lo < hi) { int mid = (lo + hi) >> 1; if (S.z_f[mid] < v) lo = mid + 1; else hi = mid; }
      S.z_all[i + lo] = v; S.sig_all[i + lo] = S.sig_c[i];
    }
    {
      const float v = S.z_f[i];
      int lo = 0, hi = 64;                       // count of z_c <= v (coarse first on ties)
      while (lo < hi) { int mid = (lo + hi) >> 1; if (S.z_c[mid] <= v) lo = mid + 1; else hi = mid; }
      S.z_all[lo + i] = v; S.sig_all[lo + i] = S.sig_f[i];
    }
  }
  __syncthreads();

  // ---- final compositing weights over 128 samples (4/lane prefix-product) ----
  float wsum = 0.f;
  {
    float al[4], qv[4];
#pragma unroll
    for (int e = 0; e < 4; ++e) {
      const int i = lane * 4 + e;
      const float zi = S.z_all[i];
      const float dd = (i < NT - 1) ? (S.z_all[i + 1] - zi) : sample_dist;
      const float aa = 1.f - expf(-dd * S.sig_all[i]);
      al[e] = aa; qv[e] = 1.f - aa + 1e-15f;
    }
    float inc = qv[0] * qv[1] * qv[2] * qv[3];
#pragma unroll
    for (int dlt = 1; dlt < 32; dlt <<= 1) {
      float up = __shfl_up(inc, dlt);
      if (lane >= dlt) inc *= up;
    }
    float T = __shfl_up(inc, 1);
    if (lane == 0) T = 1.f;
#pragma unroll
    for (int e = 0; e < 4; ++e) {
      const int i = lane * 4 + e;
      const float w = al[e] * T;
      S.w_all[i] = w;
      wsum += w;
      T *= qv[e];
    }
    wsum += __shfl_xor(wsum, 1);
    wsum += __shfl_xor(wsum, 2);
    wsum += __shfl_xor(wsum, 4);
    wsum += __shfl_xor(wsum, 8);
    wsum += __shfl_xor(wsum, 16);
  }
  __syncthreads();

  // ---- color MLP (8 tiles x 2 chained WMMAs x 4 N-tiles) + weighted accumulate ----
  float accR = 0.f, accG = 0.f, accB = 0.f;
  for (int b = 0; b < NT; b += 16)
    color_tile(S.z_all, S.w_all, b, ox, oy, oz, dxr, dyr, dzr,
               bC0, bC1, wcr, wcg, wcb, b2r, b2g, b2b, accR, accG, accB, lane);
  accR += __shfl_xor(accR, 16);
  accG += __shfl_xor(accG, 16);
  accB += __shfl_xor(accB, 16);

  if (valid && lane == 0) {
    const float bg = 1.f - wsum;                 // white background
    out[rid * 3 + 0] = accR + bg;
    out[rid * 3 + 1] = accG + bg;
    out[rid * 3 + 2] = accB + bg;
  }
}

extern "C" void kernel_launch(void* const* d_in, const int* in_sizes, int n_in,
                              void* d_out, int out_size, void* d_ws, size_t ws_size,
                              hipStream_t stream) {
  const float* rays_o  = (const float*)d_in[0];
  const float* rays_d  = (const float*)d_in[1];
  const float* W1      = (const float*)d_in[2];
  const float* b1      = (const float*)d_in[3];
  const float* w_sigma = (const float*)d_in[4];
  const float* b_sigma = (const float*)d_in[5];
  const float* Wc1     = (const float*)d_in[6];
  const float* bc1     = (const float*)d_in[7];
  const float* Wc2     = (const float*)d_in[8];
  const float* bc2     = (const float*)d_in[9];
  float* out = (float*)d_out;

  const int R = in_sizes[0] / 3;                 // total rays
  const int blocks = (R + WPB - 1) / WPB;
  nerf_fused_kernel<<<blocks, WPB * 32, 0, stream>>>(
      rays_o, rays_d, W1, b1, w_sigma, b_sigma, Wc1, bc1, Wc2, bc2, out, R);
}